// SSIM3DLoss_30932354465875
// MI455X (gfx1250) — compile-verified
//
#include <hip/hip_runtime.h>
#include <hip/hip_bf16.h>

typedef __attribute__((ext_vector_type(2))) float v2f;
typedef __attribute__((ext_vector_type(8))) float v8f;

#define NSIDE 256
#define NVOX  ((size_t)16777216)   // 256^3
#define INV11 (1.0f / 11.0f)
#define C1_SSIM 1.0e-4f            // 0.01^2
#define C2_SSIM 9.0e-4f            // 0.03^2

// ---------------------------------------------------------------------------
// Pass 1: box filter along W (contiguous axis) as banded matmul via WMMA.
// Treat x,y as 65536 x 256 matrices. out(r,w) = sum_k in(r,k) * F(k,w),
// F(k,w) = 1/11 if |k-w|<=5 and 0<=k<256 else 0 (zero padding).
// One wave per (16-row group, 16-wide w tile). 5 channels share the B (filter) tile.
// ---------------------------------------------------------------------------
__global__ void ssim3d_pass1_w(const float* __restrict__ x,
                               const float* __restrict__ y,
                               float* __restrict__ vols) {
  const int lane  = threadIdx.x & 31;
  const int wave  = (int)((blockIdx.x * blockDim.x + threadIdx.x) >> 5);
  const int g     = wave >> 4;          // row group 0..4095
  const int wt    = wave & 15;          // w tile   0..15
  const int row0  = g << 4;
  const int w0    = wt << 4;
  const int m     = lane & 15;          // A row / D column index
  const int khalf = lane >> 4;          // 0 or 1
  const int wl    = w0 + m;             // B column (w) for this lane
  const size_t rowbase = (size_t)(row0 + m) * NSIDE;

  v8f accX{}, accY{}, accXX{}, accYY{}, accXY{};

#pragma unroll
  for (int t7 = 0; t7 < 7; ++t7) {
    const int k0 = w0 - 5 + 4 * t7;     // k tile base (band start)
    const int ka = k0 + 2 * khalf;      // first k this lane supplies
    // clamp address into bounds; B carries the zero for OOB k
    int kc = ka;
    if (kc < 0) kc = 0;
    if (kc > NSIDE - 2) kc = NSIDE - 2;

    v2f ax = *(const v2f*)(x + rowbase + (size_t)kc);
    v2f ay = *(const v2f*)(y + rowbase + (size_t)kc);
    v2f axx = ax * ax;
    v2f ayy = ay * ay;
    v2f axy = ax * ay;

    // Banded filter tile B(k, n=wl): two k's per lane (ka, ka+1)
    v2f b;
    int d0 = ka - wl;     if (d0 < 0) d0 = -d0;
    int d1 = ka + 1 - wl; if (d1 < 0) d1 = -d1;
    b.x = (d0 <= 5 && (unsigned)ka       < (unsigned)NSIDE) ? INV11 : 0.0f;
    b.y = (d1 <= 5 && (unsigned)(ka + 1) < (unsigned)NSIDE) ? INV11 : 0.0f;

    accX  = __builtin_amdgcn_wmma_f32_16x16x4_f32(false, ax,  false, b, (short)0, accX,  false, false);
    accY  = __builtin_amdgcn_wmma_f32_16x16x4_f32(false, ay,  false, b, (short)0, accY,  false, false);
    accXX = __builtin_amdgcn_wmma_f32_16x16x4_f32(false, axx, false, b, (short)0, accXX, false, false);
    accYY = __builtin_amdgcn_wmma_f32_16x16x4_f32(false, ayy, false, b, (short)0, accYY, false, false);
    accXY = __builtin_amdgcn_wmma_f32_16x16x4_f32(false, axy, false, b, (short)0, accXY, false, false);
  }

  // D layout: VGPR r -> row (row0 + r + 8*khalf), column w0 + (lane&15)
  const int srow0 = row0 + khalf * 8;
  const int sw    = w0 + m;
#pragma unroll
  for (int r = 0; r < 8; ++r) {
    const size_t o = (size_t)(srow0 + r) * NSIDE + (size_t)sw;
    vols[0 * NVOX + o] = accX[r];
    vols[1 * NVOX + o] = accY[r];
    vols[2 * NVOX + o] = accXX[r];
    vols[3 * NVOX + o] = accYY[r];
    vols[4 * NVOX + o] = accXY[r];
  }
}

// ---------------------------------------------------------------------------
// Pass 2: box filter along H (stride 256), in place. Each thread owns one
// (d, w) pillar; block b handles slab d = b, thread t handles w = t.
// Sliding window sum with an 11-deep ring buffer in LDS (dynamic index).
// Writes trail reads by 5 rows within the pillar -> in-place safe.
// ---------------------------------------------------------------------------
__global__ void ssim3d_pass2_h(float* vols) {
  const int t = threadIdx.x;          // w
  const int d = blockIdx.x;           // slab
  __shared__ float ring[11 * 5 * 256];

  // zero own ring slots (thread t only ever touches indices == t mod 256)
  for (int s = t; s < 11 * 5 * 256; s += 256) ring[s] = 0.0f;

  float sum[5] = {0.f, 0.f, 0.f, 0.f, 0.f};
  const size_t base = (size_t)d * (NSIDE * NSIDE) + (size_t)t;

  for (int j = 0; j <= 260; ++j) {
    const int slot = j % 11;
    const bool inb = (j < NSIDE);
    const size_t rowoff = base + (size_t)(inb ? j : 0) * NSIDE;
#pragma unroll
    for (int c = 0; c < 5; ++c) {
      float v = 0.0f;
      if (inb) v = vols[(size_t)c * NVOX + rowoff];
      const int ra = slot * 1280 + c * 256 + t;
      const float old = ring[ra];
      ring[ra] = v;
      sum[c] += v - old;
    }
    if (j >= 5) {
      const size_t o = base + (size_t)(j - 5) * NSIDE;
#pragma unroll
      for (int c = 0; c < 5; ++c) vols[(size_t)c * NVOX + o] = sum[c] * INV11;
    }
  }
}

// ---------------------------------------------------------------------------
// Pass 3: box filter along D (stride 65536), fused SSIM map + reduction.
// Read-only on vols, so D is chunked (4 chunks of 64 with 10-step warmup).
// blockIdx: [chunk(2b) | h(8b)]; thread t owns w = t. Deterministic LDS
// tree reduction -> one partial per block.
// ---------------------------------------------------------------------------
__global__ void ssim3d_pass3_d(const float* __restrict__ vols,
                               float* __restrict__ partials) {
  const int t  = threadIdx.x;               // w
  const int h  = blockIdx.x & 255;
  const int cc = blockIdx.x >> 8;           // d chunk 0..3
  __shared__ float ring[11 * 5 * 256];
  __shared__ float red[256];

  for (int s = t; s < 11 * 5 * 256; s += 256) ring[s] = 0.0f;

  float sum[5] = {0.f, 0.f, 0.f, 0.f, 0.f};
  float acc = 0.0f;
  const size_t base = (size_t)h * NSIDE + (size_t)t;
  const int j0 = 64 * cc - 5;

  for (int k = 0; k <= 73; ++k) {
    const int j = j0 + k;
    const int slot = k % 11;
    const bool inb = ((unsigned)j < (unsigned)NSIDE);
    const size_t off = base + (size_t)(inb ? j : 0) * (NSIDE * NSIDE);
#pragma unroll
    for (int c = 0; c < 5; ++c) {
      float v = 0.0f;
      if (inb) v = vols[(size_t)c * NVOX + off];
      const int ra = slot * 1280 + c * 256 + t;
      const float old = ring[ra];
      ring[ra] = v;
      sum[c] += v - old;
    }
    if (k >= 10) {  // output index i = j-5 in [64*cc, 64*cc+63]
      const float mu_x = sum[0] * INV11;
      const float mu_y = sum[1] * INV11;
      const float sxx  = sum[2] * INV11;
      const float syy  = sum[3] * INV11;
      const float sxy  = sum[4] * INV11;
      const float mu_xy = mu_x * mu_y;
      const float mx2 = mu_x * mu_x;
      const float my2 = mu_y * mu_y;
      const float sigx  = sxx - mx2;
      const float sigy  = syy - my2;
      const float sigxy = sxy - mu_xy;
      const float num = (2.0f * mu_xy + C1_SSIM) * (2.0f * sigxy + C2_SSIM);
      const float den = (mx2 + my2 + C1_SSIM) * (sigx + sigy + C2_SSIM);
      acc += num / den;
    }
  }

  __syncthreads();
  red[t] = acc;
  __syncthreads();
#pragma unroll
  for (int s = 128; s > 0; s >>= 1) {
    if (t < s) red[t] += red[t + s];
    __syncthreads();
  }
  if (t == 0) partials[blockIdx.x] = red[0];
}

// ---------------------------------------------------------------------------
// Pass 4: deterministic final reduction of 1024 partials, loss = 1 - mean.
// ---------------------------------------------------------------------------
__global__ void ssim3d_pass4_final(const float* __restrict__ partials,
                                   float* __restrict__ out) {
  __shared__ float red[256];
  const int t = threadIdx.x;
  float a = 0.0f;
#pragma unroll
  for (int i = 0; i < 4; ++i) a += partials[t + 256 * i];
  red[t] = a;
  __syncthreads();
#pragma unroll
  for (int s = 128; s > 0; s >>= 1) {
    if (t < s) red[t] += red[t + s];
    __syncthreads();
  }
  if (t == 0) out[0] = 1.0f - red[0] * (1.0f / 16777216.0f);
}

// ---------------------------------------------------------------------------
extern "C" void kernel_launch(void* const* d_in, const int* in_sizes, int n_in,
                              void* d_out, int out_size, void* d_ws, size_t ws_size,
                              hipStream_t stream) {
  const float* x = (const float*)d_in[0];
  const float* y = (const float*)d_in[1];
  // d_in[2] = window_size (== 11, constant per reference)
  float* vols     = (float*)d_ws;            // 5 volumes of 256^3 floats
  float* partials = vols + 5 * NVOX;         // 1024 floats
  float* out      = (float*)d_out;

  ssim3d_pass1_w<<<8192, 256, 0, stream>>>(x, y, vols);
  ssim3d_pass2_h<<<256, 256, 0, stream>>>(vols);
  ssim3d_pass3_d<<<1024, 256, 0, stream>>>(vols, partials);
  ssim3d_pass4_final<<<1, 256, 0, stream>>>(partials, out);
}